// Net_87076166960184
// MI455X (gfx1250) — compile-verified
//
#include <hip/hip_runtime.h>

#define NPTS   4096
#define NB     16
#define KNN    32
#define QPB    128                 // queries per block (8 waves x 16)
#define NCHUNK (NPTS / QPB)        // 32 blocks per batch
#define INC    30
#define EMB    32
#define PAD    8                   // replicated pad for the +8 row-shift

typedef __attribute__((ext_vector_type(2))) float v2f;
typedef __attribute__((ext_vector_type(8))) float v8f;

// D = A(16x4,f32) * B(4x16,f32) + C(16x16,f32), wave32 striped operands.
static __device__ __forceinline__ v8f wmma16x16x4(v2f a, v2f b, v8f c) {
  return __builtin_amdgcn_wmma_f32_16x16x4_f32(
      /*neg_a=*/false, a, /*neg_b=*/false, b,
      /*c_mod=*/(short)0, c, /*reuse_a=*/false, /*reuse_b=*/false);
}

// ---------------------------------------------------------------------------
// Kernel 1: fused KNN + top-32 + 30-ch features + per-block max partials.
//   A = neighbor tile (n_x, n_y, n_z, |n|^2), B = query tile (-2q, 1)
//   => D[m,q] = |n|^2 - 2 n.q  (ranking score straight from the WMMA)
// Per unrolled step: 4 WMMAs over neighbor rows cb+{0,8,16,24}+lm, so lane
// (hf,lm) streams candidates idx = cb + hf*8 + j, j in [0,32) -- each lane
// owns one full query column, no transpose, no cross-lane merge.
// Grid: (NCHUNK, NB), 256 threads.
// ---------------------------------------------------------------------------
__global__ __launch_bounds__(256, 1) void knn_feat_kernel(
    const float* __restrict__ x, float* __restrict__ partial) {
  __shared__ float2 Pxy[NPTS + PAD];            // (x, y)
  __shared__ float2 Pzq[NPTS + PAD];            // (z, |p|^2)
  __shared__ __align__(16) float cand[256][32]; // slow-path spill (128B/lane)
  __shared__ float wred[8][INC];

  const int tid  = threadIdx.x;
  const int b    = blockIdx.y;
  const int wv   = tid >> 5;
  const int lane = tid & 31;
  const int hf   = lane >> 4;     // 0: supplies K0,K1 / rows +0..7 of D
  const int lm   = lane & 15;
  const int hf8  = hf << 3;

  // Stage the batch point cloud as interleaved float2 pairs (b128 loads).
  const float*  xb  = x + (size_t)b * NPTS * 3;
  const float4* xb4 = (const float4*)xb;
  for (int g = tid; g < NPTS / 4; g += 256) {   // 4 points per group
    float4 f0 = xb4[g * 3 + 0], f1 = xb4[g * 3 + 1], f2 = xb4[g * 3 + 2];
    int i = g * 4;
    Pxy[i + 0] = make_float2(f0.x, f0.y);
    Pzq[i + 0] = make_float2(f0.z, f0.x * f0.x + f0.y * f0.y + f0.z * f0.z);
    Pxy[i + 1] = make_float2(f0.w, f1.x);
    Pzq[i + 1] = make_float2(f1.y, f0.w * f0.w + f1.x * f1.x + f1.y * f1.y);
    Pxy[i + 2] = make_float2(f1.z, f1.w);
    Pzq[i + 2] = make_float2(f2.x, f1.z * f1.z + f1.w * f1.w + f2.x * f2.x);
    Pxy[i + 3] = make_float2(f2.y, f2.z);
    Pzq[i + 3] = make_float2(f2.w, f2.y * f2.y + f2.z * f2.z + f2.w * f2.w);
  }
  if (tid < PAD) {                              // replicate points 0..7
    float px = xb[tid * 3 + 0], py = xb[tid * 3 + 1], pz = xb[tid * 3 + 2];
    Pxy[NPTS + tid] = make_float2(px, py);
    Pzq[NPTS + tid] = make_float2(pz, px * px + py * py + pz * pz);
  }
  __syncthreads();

  // This wave owns queries qb..qb+15; lane's query column is qb+lm.
  const int qb = blockIdx.x * QPB + wv * 16;
  const int q  = qb + lm;
  const float2 qxy = Pxy[q];
  const float2 qzq = Pzq[q];

  v2f bq;                                   // 4x16 B-tile (query), invariant
  bq.x = hf ? (-2.0f * qzq.x) : (-2.0f * qxy.x);   // K2 : K0
  bq.y = hf ? 1.0f            : (-2.0f * qxy.y);   // K3 : K1

  const float2* pA = hf ? Pzq : Pxy;        // A operand source for this lane

  // Per-lane private top-32 (unsorted + cached max), static indexing only.
  float bd[KNN]; int bi[KNN];
#pragma unroll
  for (int t = 0; t < KNN; ++t) { bd[t] = 3.0e38f; bi[t] = 0; }
  float worst = 3.0e38f;

  float* cl = &cand[tid][0];
  const float2* pc = pA + lm;
  float2 c0 = pc[0], c1 = pc[8], c2 = pc[16], c3 = pc[24];
  for (int it = 0; it < NPTS / 32; ++it) {
    const int cb = it << 5;
    // software prefetch of the next step's operands (overlaps the WMMAs)
    const float2* pn = pc + 32;
    float2 n0 = pn[0], n1 = pn[8], n2 = pn[16], n3 = pn[24];

    v2f a0 = {c0.x, c0.y}, a1 = {c1.x, c1.y};
    v2f a2 = {c2.x, c2.y}, a3 = {c3.x, c3.y};
    v8f cz = {};
    v8f w0 = wmma16x16x4(a0, bq, cz);   // rows cb +  0 + hf8 + v
    v8f w1 = wmma16x16x4(a1, bq, cz);   // rows cb +  8 + hf8 + v
    v8f w2 = wmma16x16x4(a2, bq, cz);   // rows cb + 16 + hf8 + v
    v8f w3 = wmma16x16x4(a3, bq, cz);   // rows cb + 24 + hf8 + v

    // Quick reject: min-tree over all 32 candidates, one branch per step.
    float mA = fminf(fminf(fminf(w0[0], w0[1]), fminf(w0[2], w0[3])),
                     fminf(fminf(w0[4], w0[5]), fminf(w0[6], w0[7])));
    float mB = fminf(fminf(fminf(w1[0], w1[1]), fminf(w1[2], w1[3])),
                     fminf(fminf(w1[4], w1[5]), fminf(w1[6], w1[7])));
    float mC = fminf(fminf(fminf(w2[0], w2[1]), fminf(w2[2], w2[3])),
                     fminf(fminf(w2[4], w2[5]), fminf(w2[6], w2[7])));
    float mD = fminf(fminf(fminf(w3[0], w3[1]), fminf(w3[2], w3[3])),
                     fminf(fminf(w3[4], w3[5]), fminf(w3[6], w3[7])));
    float tmin = fminf(fminf(mA, mB), fminf(mC, mD));

    if (tmin < worst) {                    // rare after warm-up
      // Spill to per-lane LDS line (merged b128 stores) so the insert loop
      // stays non-unrolled: one copy of the 32-slot replace-max code.
#pragma unroll
      for (int v = 0; v < 8; ++v) {
        cl[v] = w0[v]; cl[8 + v] = w1[v]; cl[16 + v] = w2[v]; cl[24 + v] = w3[v];
      }
      const int cbh = cb + hf8;            // candidate idx = cbh + j
#pragma clang loop unroll(disable)
      for (int j = 0; j < 32; ++j) {
        float s = cl[j];
        if (s < worst) {
          int idx = cbh + j;
          float m = bd[0]; int arg = 0;
#pragma unroll
          for (int t = 1; t < KNN; ++t) { bool g = bd[t] > m; m = g ? bd[t] : m; arg = g ? t : arg; }
#pragma unroll
          for (int t = 0; t < KNN; ++t) {
            bool w = (t == arg);
            bd[t] = w ? s : bd[t];
            bi[t] = w ? idx : bi[t];
          }
          float nw = bd[0];
#pragma unroll
          for (int t = 1; t < KNN; ++t) nw = fmaxf(nw, bd[t]);
          worst = nw;
        }
      }
    }
    pc = pn; c0 = n0; c1 = n1; c2 = n2; c3 = n3;
  }

  // ---- per-point features over the selected 32 neighbors ----
  const float qx = qxy.x, qy = qxy.y, qz = qzq.x;
  float sx = 0, sy = 0, sz = 0, qsx = 0, qsy = 0, qsz = 0;
  float mxx = -3.0e38f, mxy = -3.0e38f, mxz = -3.0e38f;
  float mnx =  3.0e38f, mny =  3.0e38f, mnz =  3.0e38f;
#pragma unroll
  for (int t = 0; t < KNN; ++t) {
    int n = bi[t];                       // may hit pad alias: valid coords
    float2 nxy = Pxy[n]; float nz = Pzq[n].x;
    float rx = nxy.x - qx, ry = nxy.y - qy, rz = nz - qz;
    sx += rx; sy += ry; sz += rz;
    qsx = fmaf(rx, rx, qsx); qsy = fmaf(ry, ry, qsy); qsz = fmaf(rz, rz, qsz);
    mxx = fmaxf(mxx, rx); mxy = fmaxf(mxy, ry); mxz = fmaxf(mxz, rz);
    mnx = fminf(mnx, rx); mny = fminf(mny, ry); mnz = fminf(mnz, rz);
  }
  const float kinv = 1.0f / (float)KNN;
  float mux = sx * kinv, muy = sy * kinv, muz = sz * kinv;
  float msx = qsx * kinv, msy = qsy * kinv, msz = qsz * kinv;
  float stx = sqrtf(fmaxf(msx - mux * mux, 0.0f));
  float sty = sqrtf(fmaxf(msy - muy * muy, 0.0f));
  float stz = sqrtf(fmaxf(msz - muz * muz, 0.0f));
  float mn  = sqrtf(mux * mux + muy * muy + muz * muz);
  float inv = 1.0f / (mn + 1e-8f);
  float ux = mux * inv, uy = muy * inv, uz = muz * inv;
  float cxx = qy * uz - qz * uy;
  float cyy = qz * ux - qx * uz;
  float crr = qx * uy - qy * ux;

  float f[INC] = { qx, qy, qz,  mux, muy, muz,  mxx, mxy, mxz,
                   mnx, mny, mnz,  stx, sty, stz,
                   qx - mux, qy - muy, qz - muz,  ux, uy, uz,  cxx, cyy, crr,
                   fmaxf(mxx * mxx, mnx * mnx), fmaxf(mxy * mxy, mny * mny),
                   fmaxf(mxz * mxz, mnz * mnz), msx, msy, msz };

  // max-reduce across the block (wave32 shuffle tree, then cross-wave via LDS)
#pragma unroll
  for (int ch = 0; ch < INC; ++ch) {
    float v = f[ch];
#pragma unroll
    for (int off = 16; off > 0; off >>= 1)
      v = fmaxf(v, __shfl_xor(v, off, 32));
    if (lane == 0) wred[wv][ch] = v;
  }
  __syncthreads();
  if (tid < INC) {
    float m = wred[0][tid];
#pragma unroll
    for (int w = 1; w < 8; ++w) m = fmaxf(m, wred[w][tid]);
    partial[((size_t)b * NCHUNK + blockIdx.x) * INC + tid] = m;
  }
}

// ---------------------------------------------------------------------------
// Kernel 2: final max-pool over chunk partials + [16,30]@W^T[30,32]+b via
//           chained f32 16x16x4 WMMAs.  1 block, 64 threads (2 waves).
// ---------------------------------------------------------------------------
__global__ __launch_bounds__(64, 1) void pool_linear_kernel(
    const float* __restrict__ partial, const float* __restrict__ W,
    const float* __restrict__ bias, float* __restrict__ out) {
  __shared__ float pooled[NB][32];   // K padded 30 -> 32
  __shared__ float Wsh[EMB][32];
  const int tid = threadIdx.x;

  for (int i = tid; i < NB * 32; i += 64)  ((float*)pooled)[i] = 0.0f;
  for (int i = tid; i < EMB * 32; i += 64) ((float*)Wsh)[i]    = 0.0f;
  __syncthreads();
  for (int i = tid; i < NB * INC; i += 64) {
    int bb = i / INC, ch = i % INC;
    float m = -3.0e38f;
    for (int c = 0; c < NCHUNK; ++c)
      m = fmaxf(m, partial[((size_t)bb * NCHUNK + c) * INC + ch]);
    pooled[bb][ch] = m;
  }
  for (int i = tid; i < EMB * INC; i += 64) Wsh[i / INC][i % INC] = W[i];
  __syncthreads();

  const int wv = tid >> 5, lane = tid & 31, hf = lane >> 4, lm = lane & 15;
  const int nb = wv * 16;            // output-column tile (emb 0-15 / 16-31)
  v8f acc = {};
#pragma unroll
  for (int kt = 0; kt < 8; ++kt) {   // K = 32 in steps of 4
    int k0 = kt * 4 + hf * 2;
    v2f a;  a.x  = pooled[lm][k0];     a.y  = pooled[lm][k0 + 1];
    v2f bt; bt.x = Wsh[nb + lm][k0];   bt.y = Wsh[nb + lm][k0 + 1];
    acc = wmma16x16x4(a, bt, acc);
  }
  float bv = bias[nb + lm];
#pragma unroll
  for (int v = 0; v < 8; ++v) {
    int m = hf * 8 + v;                         // D row; col = nb + lm
    out[m * EMB + nb + lm] = acc[v] + bv;
  }
}

// ---------------------------------------------------------------------------
extern "C" void kernel_launch(void* const* d_in, const int* in_sizes, int n_in,
                              void* d_out, int out_size, void* d_ws, size_t ws_size,
                              hipStream_t stream) {
  const float* x    = (const float*)d_in[0];   // [16,4096,3]
  const float* W    = (const float*)d_in[1];   // [32,30]
  const float* bias = (const float*)d_in[2];   // [32]
  float* out     = (float*)d_out;              // [16,32]
  float* partial = (float*)d_ws;               // [16][32][30] floats (60 KB)

  knn_feat_kernel<<<dim3(NCHUNK, NB), 256, 0, stream>>>(x, partial);
  pool_linear_kernel<<<1, 64, 0, stream>>>(partial, W, bias, out);
}